// TripletLoss_12343736009507
// MI455X (gfx1250) — compile-verified
//
#include <hip/hip_runtime.h>
#include <math.h>

// ---------------------------------------------------------------------------
// Batch-hard triplet loss, fused: Gram via V_WMMA_F32_16X16X4_F32 + mining.
// B = 4096 rows, D = 128 dims, fp32 throughout (matches reference numerics).
// ---------------------------------------------------------------------------

typedef __attribute__((ext_vector_type(2))) float v2f;
typedef __attribute__((ext_vector_type(8))) float v8f;

#define BN      4096
#define DK      128
#define TMARGIN 0.3f

#define PANEL      64     // columns staged in LDS per iteration
#define LDS_STRIDE 132    // 128 + 4 pad -> conflict-free ds_load_b64 B reads

// Workspace layout (floats): [0]=loss sum, [1]=valid count, [2..2+BN)=sq norms
__global__ __launch_bounds__(256) void sqnorm_kernel(const float* __restrict__ X,
                                                     float* __restrict__ ws) {
    int r = blockIdx.x * blockDim.x + threadIdx.x;
    if (r == 0) { ws[0] = 0.0f; ws[1] = 0.0f; }
    if (r < BN) {
        const float4* x4 = (const float4*)(X + (size_t)r * DK);
        float s = 0.0f;
#pragma unroll
        for (int i = 0; i < DK / 4; ++i) {
            float4 v = x4[i];
            s += v.x * v.x + v.y * v.y + v.z * v.z + v.w * v.w;
        }
        ws[2 + r] = s;
    }
}

__global__ __launch_bounds__(256) void triplet_kernel(const float* __restrict__ X,
                                                      const int* __restrict__ labels,
                                                      float* __restrict__ ws) {
    __shared__ float lds[PANEL * LDS_STRIDE];

    const float* sq = ws + 2;
    const int tid  = threadIdx.x;
    const int wave = tid >> 5;
    const int lane = tid & 31;
    const int half = lane >> 4;    // K-pair half for A/B fragments
    const int lm   = lane & 15;    // row (A) / col (B) within tile
    const int rowbase = blockIdx.x * 128 + wave * 16;

    // --- A panel: 16 rows x 128 K held as 32 WMMA f32 16x16x4 fragments ---
    // Lane l<16 holds row l, K = {4k, 4k+1}; lane l+16 holds row l, K = {4k+2, 4k+3}.
    v2f a[32];
    {
        const float* xr = X + (size_t)(rowbase + lm) * DK + 2 * half;
#pragma unroll
        for (int kk = 0; kk < 32; ++kk)
            a[kk] = *(const v2f*)(xr + kk * 4);
    }

    // Per-lane metadata for the 8 rows this lane's accumulator slots cover:
    // C/D layout: vgpr v <-> row (v + 8*half).
    float sqr[8];
    int   labr[8];
#pragma unroll
    for (int v = 0; v < 8; ++v) {
        int rg  = rowbase + 8 * half + v;
        sqr[v]  = sq[rg];
        labr[v] = labels[rg];
    }

    float maxpos[8], minneg[8];
#pragma unroll
    for (int v = 0; v < 8; ++v) { maxpos[v] = -INFINITY; minneg[v] = INFINITY; }

    const float4* x4 = (const float4*)X;

    for (int panel = 0; panel < BN / PANEL; ++panel) {
        const int colpanel = panel * PANEL;
        __syncthreads();
        // Stage PANEL columns x 128 K into LDS (2048 float4s / 256 threads).
        for (int i = tid; i < PANEL * (DK / 4); i += 256) {
            int c  = i >> 5;       // column within panel
            int kq = i & 31;       // float4 index along K
            float4 val = x4[(size_t)(colpanel + c) * (DK / 4) + kq];
            *(float4*)&lds[c * LDS_STRIDE + kq * 4] = val;
        }
        __syncthreads();

#pragma unroll
        for (int t = 0; t < PANEL / 16; ++t) {
            const int col  = colpanel + t * 16 + lm;
            const float sqc = sq[col];
            const int   labc = labels[col];

            v8f acc = {};
            const float* bp = &lds[(t * 16 + lm) * LDS_STRIDE + 2 * half];
#pragma unroll
            for (int kk = 0; kk < 32; ++kk) {
                v2f b = *(const v2f*)(bp + kk * 4);
                acc = __builtin_amdgcn_wmma_f32_16x16x4_f32(
                    false, a[kk], false, b, (short)0, acc, false, false);
            }

            // Fused mining on the 16x16 Gram tile.
#pragma unroll
            for (int v = 0; v < 8; ++v) {
                float g  = acc[v];
                float d2 = sqr[v] + sqc - 2.0f * g;
                float d  = sqrtf(fmaxf(d2, 0.0f));
                int rg   = rowbase + 8 * half + v;
                bool same  = (labr[v] == labc);
                bool ispos = same && (rg != col);
                bool isneg = !same;
                maxpos[v] = ispos ? fmaxf(maxpos[v], d) : maxpos[v];
                minneg[v] = isneg ? fminf(minneg[v], d) : minneg[v];
            }
        }
    }

    // Reduce the 16 column-lanes per row half, then accumulate globally.
#pragma unroll
    for (int v = 0; v < 8; ++v) {
        float mp = maxpos[v];
        float mn = minneg[v];
#pragma unroll
        for (int m = 8; m >= 1; m >>= 1) {
            mp = fmaxf(mp, __shfl_xor(mp, m, 32));
            mn = fminf(mn, __shfl_xor(mn, m, 32));
        }
        if (lm == 0) {
            // distances are >= 0, so any positive hit makes mp >= 0
            bool valid = (mp >= 0.0f) && (mn < 3.0e38f);
            float per  = valid ? fmaxf(mp - mn + TMARGIN, 0.0f) : 0.0f;
            atomicAdd(&ws[0], per);
            atomicAdd(&ws[1], valid ? 1.0f : 0.0f);
        }
    }
}

__global__ void finalize_kernel(const float* __restrict__ ws, float* __restrict__ out) {
    out[0] = ws[0] / fmaxf(ws[1], 1.0f);
}

extern "C" void kernel_launch(void* const* d_in, const int* in_sizes, int n_in,
                              void* d_out, int out_size, void* d_ws, size_t ws_size,
                              hipStream_t stream) {
    const float* X      = (const float*)d_in[0];   // [4096, 128] fp32
    const int*   labels = (const int*)d_in[1];     // [4096] int32
    float*       out    = (float*)d_out;           // scalar
    float*       ws     = (float*)d_ws;            // >= (2 + 4096) floats

    sqnorm_kernel<<<BN / 256, 256, 0, stream>>>(X, ws);
    triplet_kernel<<<BN / 128, 256, 0, stream>>>(X, labels, ws);
    finalize_kernel<<<1, 1, 0, stream>>>(ws, out);
}